// TrainableLayer_65670049956245
// MI455X (gfx1250) — compile-verified
//
#include <hip/hip_runtime.h>
#include <hip/hip_bf16.h>
#include <math.h>

// Problem constants from the reference
#define N_NODES 256
#define BATCH   4096
#define F_SIZE  64
#define T_SIZE  64

// Block tiling: 256 threads = 8 waves; each wave computes 16 rows x 64 cols.
#define BM 128          // rows of the batch handled per block (8 waves * 16 rows)

// LDS weight layout: K-pairs interleaved per column so a B fragment is one
// contiguous 8B ds_load_b64:  sW[(k>>1)*KP_STRIDE + 2*col + (k&1)]
// Pair stride padded 128 -> 160 floats so lanes 0-15 (K offset 0, banks 0-31)
// and lanes 16-31 (K offset 2, banks 32-63) never collide.
#define KP_STRIDE 160

typedef __attribute__((ext_vector_type(2))) float v2f;
typedef __attribute__((ext_vector_type(8))) float v8f;

__global__ __launch_bounds__(256)
void node_gemm_sigmoid_f32wmma(const float* __restrict__ x,
                               const float* __restrict__ W,
                               const float* __restrict__ mask,
                               const float* __restrict__ bias,
                               float* __restrict__ out)
{
    __shared__ float sW[(F_SIZE / 2) * KP_STRIDE];   // 32 k-pairs * 160 = 20 KB
    __shared__ float sB[T_SIZE];

    const int node  = blockIdx.x;          // 0..255
    const int btile = blockIdx.y;          // 0..31
    const int tid   = threadIdx.x;         // 0..255
    const int wave  = tid >> 5;            // 0..7 (wave32)
    const int lane  = tid & 31;
    const int l16   = lane & 15;
    const int hiHalf = lane >> 4;          // 0 for lanes 0-15, 1 for 16-31

    // ---- Stage W * mask into LDS in WMMA B-fragment order ----
    {
        const float* Wn = W    + (size_t)node * (F_SIZE * T_SIZE);
        const float* Mn = mask + (size_t)node * (F_SIZE * T_SIZE);
        #pragma unroll
        for (int i = tid; i < F_SIZE * T_SIZE; i += 256) {
            const int k   = i >> 6;        // 0..63
            const int col = i & 63;        // 0..63
            sW[(k >> 1) * KP_STRIDE + 2 * col + (k & 1)] = Wn[i] * Mn[i];
        }
        if (tid < T_SIZE)
            sB[tid] = bias[(size_t)node * T_SIZE + tid];
    }
    __syncthreads();

    // ---- A (x) addressing: each wave owns rows [rowBase, rowBase+16) ----
    const int rowBase = btile * BM + wave * 16;
    const float* aRow = x + ((size_t)node * BATCH + (size_t)(rowBase + l16)) * F_SIZE;
    const int kHalf = hiHalf * 2;          // lanes 0-15 -> K offset 0; 16-31 -> K offset 2

    // B fragment base: k-pair row = 2*kk + hiHalf, element = 2*col
    const float* bBase = &sW[(size_t)hiHalf * KP_STRIDE + 2 * l16];

    v8f acc0 = {}, acc1 = {}, acc2 = {}, acc3 = {};

    // ---- K loop: 16 steps of V_WMMA_F32_16X16X4_F32 per output tile ----
    #pragma unroll
    for (int kk = 0; kk < 16; ++kk) {
        const int k0 = 4 * kk + kHalf;     // even -> 8B-aligned float2 load

        // A fragment: x[row, k0], x[row, k0+1] — streaming, non-temporal b64
        v2f a = __builtin_nontemporal_load((const v2f*)(aRow + k0));

        // B fragments: one contiguous ds_load_b64 each (immediate offsets)
        const float* bk = bBase + (size_t)(2 * kk) * KP_STRIDE;
        v2f b0 = *(const v2f*)(bk + 0);    // cols  0..15
        v2f b1 = *(const v2f*)(bk + 32);   // cols 16..31
        v2f b2 = *(const v2f*)(bk + 64);   // cols 32..47
        v2f b3 = *(const v2f*)(bk + 96);   // cols 48..63

        // D = A x B + C   (8 args: neg_a, A, neg_b, B, c_mod, C, reuse_a, reuse_b)
        acc0 = __builtin_amdgcn_wmma_f32_16x16x4_f32(false, a, false, b0, (short)0, acc0, false, false);
        acc1 = __builtin_amdgcn_wmma_f32_16x16x4_f32(false, a, false, b1, (short)0, acc1, false, false);
        acc2 = __builtin_amdgcn_wmma_f32_16x16x4_f32(false, a, false, b2, (short)0, acc2, false, false);
        acc3 = __builtin_amdgcn_wmma_f32_16x16x4_f32(false, a, false, b3, (short)0, acc3, false, false);
    }

    // ---- Epilogue: bias + sigmoid + non-temporal store ----
    // C/D layout: VGPR i, lanes 0-15 -> M = i, N = lane; lanes 16-31 -> M = i+8, N = lane-16
    float* outBase = out + ((size_t)node * BATCH + (size_t)rowBase) * T_SIZE;
    const int rOff = hiHalf * 8;
    const float bb0 = sB[ 0 + l16];
    const float bb1 = sB[16 + l16];
    const float bb2 = sB[32 + l16];
    const float bb3 = sB[48 + l16];

    #pragma unroll
    for (int i = 0; i < 8; ++i) {
        float* rowPtr = outBase + (size_t)(i + rOff) * T_SIZE;
        float v;

        v = acc0[i] + bb0;
        __builtin_nontemporal_store(1.0f / (1.0f + __expf(-v)), rowPtr +  0 + l16);
        v = acc1[i] + bb1;
        __builtin_nontemporal_store(1.0f / (1.0f + __expf(-v)), rowPtr + 16 + l16);
        v = acc2[i] + bb2;
        __builtin_nontemporal_store(1.0f / (1.0f + __expf(-v)), rowPtr + 32 + l16);
        v = acc3[i] + bb3;
        __builtin_nontemporal_store(1.0f / (1.0f + __expf(-v)), rowPtr + 48 + l16);
    }
}

extern "C" void kernel_launch(void* const* d_in, const int* in_sizes, int n_in,
                              void* d_out, int out_size, void* d_ws, size_t ws_size,
                              hipStream_t stream) {
    const float* x    = (const float*)d_in[0];   // [256, 4096, 64]
    const float* W    = (const float*)d_in[1];   // [256, 64, 64]
    const float* mask = (const float*)d_in[2];   // [256, 64, 64]
    const float* bias = (const float*)d_in[3];   // [256, 1, 64]
    float* out = (float*)d_out;                  // [256, 4096, 64]

    dim3 grid(N_NODES, BATCH / BM, 1);           // 256 x 32 blocks
    dim3 block(256, 1, 1);                       // 8 wave32 waves
    node_gemm_sigmoid_f32wmma<<<grid, block, 0, stream>>>(x, W, mask, bias, out);
}